// SlaterDeterminant_76312978915892
// MI455X (gfx1250) — compile-verified
//
#include <hip/hip_runtime.h>
#include <cstdint>

#define NFO 32          // occupied orbitals = matrix dim
#define NOO 128         // total orbitals
#define WPB 8           // waves per 256-thread block (wave32)
#define PITCH_T 36      // dwords per transposed-Phi column in LDS (144B, 16B aligned)
#define PITCH_S 17      // dword pitch for 16x16 scratch tiles (bank stagger)

typedef float v2f __attribute__((ext_vector_type(2)));
typedef float v8f __attribute__((ext_vector_type(8)));

__device__ __forceinline__ float bc(float v, int src) {
  // wave32 broadcast via ds_bpermute (DS pipe, overlaps VALU FMAs)
  return __shfl(v, src, 32);
}

__global__ __launch_bounds__(256) void slater_det_kernel(
    const int* __restrict__ nmat, const float* __restrict__ Phi,
    float* __restrict__ out, int nb)
{
  __shared__ __align__(16) float phiT[NOO * PITCH_T];  // Phi transposed: [col][row]
  __shared__ int   idxBuf[WPB * NFO];                  // per-wave compacted indices
  __shared__ float tA[WPB][16 * PITCH_S];              // -A21  (WMMA A source)
  __shared__ float tB[WPB][16 * PITCH_S];              //  X = A11^-1 A12 (WMMA B source)
  __shared__ float tC[WPB][16 * PITCH_S];              //  A22 -> Schur S (WMMA C/D)

  const int tid    = threadIdx.x;
  const int lane   = tid & 31;
  const int wave   = tid >> 5;
  const int sample = blockIdx.x * WPB + wave;
  const bool active = sample < nb;
  const int  samp   = active ? sample : 0;

  // ---- Stage Phi (32x128 row-major) into LDS, transposed, via CDNA5 async
  //      global->LDS copy. Per-lane LDS address does the transpose scatter.
  #pragma unroll
  for (int i = 0; i < 16; ++i) {
    int e   = tid + 256 * i;          // 4096 elements / 256 threads
    int row = e >> 7;                 // 0..31
    int col = e & 127;                // 0..127
    unsigned ldsOff = (unsigned)(uintptr_t)(&phiT[col * PITCH_T + row]);
    const float* gp = Phi + e;
    asm volatile("global_load_async_to_lds_b32 %0, %1, off"
                 :: "v"(ldsOff), "v"(gp) : "memory");
  }

  // ---- Occupancy -> ascending occupied-index list (ballot + rank compaction).
  //      n is streamed once: non-temporal loads keep 64MB out of L2.
  const int* nrow = nmat + (size_t)samp * NOO;
  unsigned base = 0;
  #pragma unroll
  for (int w = 0; w < 4; ++w) {
    int occ = __builtin_nontemporal_load(&nrow[w * 32 + lane]) > 0;
    unsigned msk = (unsigned)__ballot(occ);
    if (occ) {
      int pos = (int)(base + __popc(msk & ((1u << lane) - 1u)));
      idxBuf[wave * NFO + pos] = w * 32 + lane;
    }
    base += __popc(msk);
  }
  asm volatile("" ::: "memory");
  int myCol = idxBuf[wave * NFO + lane];       // lane j owns matrix column j

  asm volatile("s_wait_asynccnt 0" ::: "memory");
  __syncthreads();                              // Phi fully staged for all waves

  // ---- Gather: lane j = column idx_j of Phi (32 contiguous floats in phiT)
  float A[32];
  {
    const float4* cp = (const float4*)(&phiT[myCol * PITCH_T]);
    #pragma unroll
    for (int r = 0; r < 8; ++r) {
      float4 v = cp[r];
      A[4*r+0] = v.x; A[4*r+1] = v.y; A[4*r+2] = v.z; A[4*r+3] = v.w;
    }
  }

  float det = 1.0f;

  // ---- Phase 1: Gauss-Jordan on rows 0..15 of [A11 | A12] (column-parallel:
  //      lanes 0..15 hold A11 cols, lanes 16..31 hold A12 cols).
  //      Accumulates det(A11) = prod(pivots); leaves X = A11^-1 A12 in
  //      registers 0..15 of lanes 16..31. Rows 16..31 (A21|A22) untouched.
  #pragma unroll
  for (int k = 0; k < 16; ++k) {
    float pv  = bc(A[k], k);
    det *= pv;
    float inv = __builtin_amdgcn_rcpf(pv);
    inv = inv * __builtin_fmaf(-pv, inv, 2.0f);   // one Newton step
    float m = A[k] * inv;                          // normalized pivot row
    #pragma unroll
    for (int r = 0; r < 16; ++r) {
      if (r == k) continue;
      float c = bc(A[r], k);                       // column-k entry of row r
      A[r] = __builtin_fmaf(-c, m, A[r]);
    }
    A[k] = m;
  }

  // ---- Dump -A21 / A22 / X into LDS tiles (layout conversion medium)
  const bool left = lane < 16;
  const int  c16  = lane & 15;
  #pragma unroll
  for (int r = 0; r < 16; ++r) {
    float v = A[16 + r];                           // rows 16..31: A21 (left) / A22 (right)
    float* dst = left ? &tA[wave][r * PITCH_S + c16]
                      : &tC[wave][r * PITCH_S + c16];
    *dst = left ? -v : v;                          // store -A21 so WMMA does C - A21*X
  }
  if (!left) {
    #pragma unroll
    for (int r = 0; r < 16; ++r)
      tB[wave][r * PITCH_S + c16] = A[r];          // X[k][n]
  }
  asm volatile("" ::: "memory");

  // ---- Load WMMA fragments per ISA layouts.
  // C/D 16x16 f32: VGPR v, lanes 0-15 -> (M=v, N=lane), lanes 16-31 -> (M=v+8).
  v8f cf;
  const int mrow = left ? 0 : 8;
  #pragma unroll
  for (int v = 0; v < 8; ++v)
    cf[v] = tC[wave][(v + mrow) * PITCH_S + c16];

#if __has_builtin(__builtin_amdgcn_wmma_f32_16x16x4_f32)
  // Schur update S = A22 + (-A21)*X as 4 chained V_WMMA_F32_16X16X4_F32 (K=16).
  // A 16x4 f32: VGPR0 = K {0 | 2}, VGPR1 = K {1 | 3}, lane%16 = M.
  // B  4x16 f32 (mirror): VGPR0 = K {0 | 2}, VGPR1 = K {1 | 3}, lane%16 = N.
  #pragma unroll
  for (int t = 0; t < 4; ++t) {
    const int kb = 4 * t + (left ? 0 : 2);
    v2f af, bf;
    af.x = tA[wave][c16 * PITCH_S + kb];
    af.y = tA[wave][c16 * PITCH_S + kb + 1];
    bf.x = tB[wave][kb * PITCH_S + c16];
    bf.y = tB[wave][(kb + 1) * PITCH_S + c16];
    cf = __builtin_amdgcn_wmma_f32_16x16x4_f32(false, af, false, bf,
                                               (short)0, cf, false, false);
  }
#else
  // Correct scalar fallback (8x16 FMAs per lane) if the builtin is absent.
  #pragma unroll
  for (int v = 0; v < 8; ++v) {
    float acc = cf[v];
    #pragma unroll
    for (int kk = 0; kk < 16; ++kk)
      acc = __builtin_fmaf(tA[wave][(v + mrow) * PITCH_S + kk],
                           tB[wave][kk * PITCH_S + c16], acc);
    cf[v] = acc;
  }
#endif

  // ---- D (C-layout) -> LDS -> column-parallel layout for the final 16x16 LU
  #pragma unroll
  for (int v = 0; v < 8; ++v)
    tC[wave][(v + mrow) * PITCH_S + c16] = cf[v];
  asm volatile("" ::: "memory");

  float S[16];
  #pragma unroll
  for (int r = 0; r < 16; ++r)
    S[r] = tC[wave][r * PITCH_S + c16];            // lanes j and j+16 mirror column j

  // ---- Phase 2: forward elimination on S, det *= prod(pivots)
  #pragma unroll
  for (int k = 0; k < 16; ++k) {
    float pv  = bc(S[k], k);
    det *= pv;
    float inv = __builtin_amdgcn_rcpf(pv);
    inv = inv * __builtin_fmaf(-pv, inv, 2.0f);
    float m = S[k] * inv;
    #pragma unroll
    for (int r = k + 1; r < 16; ++r) {
      float c = bc(S[r], k);
      S[r] = __builtin_fmaf(-c, m, S[r]);
    }
  }

  if (active && lane == 0)
    __builtin_nontemporal_store(det, &out[sample]);
}

extern "C" void kernel_launch(void* const* d_in, const int* in_sizes, int n_in,
                              void* d_out, int out_size, void* d_ws, size_t ws_size,
                              hipStream_t stream) {
  const int*   nmat = (const int*)d_in[0];    // n:   (B, 128) int32
  const float* Phi  = (const float*)d_in[1];  // Phi: (32, 128) f32
  float* out = (float*)d_out;                 // (B,) f32
  const int b = in_sizes[0] / NOO;            // 131072
  const int blocks = (b + WPB - 1) / WPB;     // one wave32 per sample
  slater_det_kernel<<<blocks, 256, 0, stream>>>(nmat, Phi, out, b);
}